// CompositeLossV3_47124381172367
// MI455X (gfx1250) — compile-verified
//
#include <hip/hip_runtime.h>

// ---------------------------------------------------------------------------
// CompositeLossV3 for MI455X (gfx1250, wave32).
// Shapes: logits[4,3,64,64,64] f32, targets[4,64,64,64] i64, skeleton i32.
// Output: 4 floats (total, lfd, lsk, lb).
//
// Memory-bound workload (~33 MB total traffic @ 23.3 TB/s). Structure:
//   1) edt_init: d2_pos/d2_neg volumes (0 or BIG) in workspace.
//   2) edt_pass x6: exact separable min-plus EDT, axis order D,H,W to match
//      the reference's float association order bit-for-bit.
//   3) reduce: one fused pass computing 26 partial sums per block
//      (focal, dice inter/union, skeleton, per-batch boundary terms),
//      deterministic LDS tree reduction (no float atomics).
//   4) finalize: chained V_WMMA_F32_16X16X4_F32 (B = ones) sums the
//      256x26 partial matrix on the matrix pipe with a fixed hardware
//      accumulation order, then computes the 4 scalars.
// Workspace: 2 * 4 MB EDT volumes + 32 KB partials = ~8.03 MB.
// ---------------------------------------------------------------------------

typedef __attribute__((ext_vector_type(2))) float v2f;
typedef __attribute__((ext_vector_type(8))) float v8f;

#define BIGF   1.0e8f
#define NVOX   (1 << 18)   // 64^3 voxels per batch
#define TVOX   (1 << 20)   // 4 * 64^3
#define NBLK   256         // reduction blocks (=> 256 partial rows)

// ---------------------------------------------------------------- EDT init
__global__ void edt_init_kernel(const long long* __restrict__ tgt,
                                float* __restrict__ pos,
                                float* __restrict__ neg) {
  int v = blockIdx.x * blockDim.x + threadIdx.x;
  if (v < TVOX) {
    bool g = (tgt[v] == 1LL);
    pos[v] = g ? 0.0f : BIGF;
    neg[v] = g ? BIGF : 0.0f;
  }
}

// ------------------------------------------------------------- EDT 1D pass
// One 64-element pencil per 64 threads (4 pencils per 256-thread block).
// base = (l / stride) * (stride*64) + (l % stride) covers W(1), H(64), D(4096).
__global__ void edt_pass_kernel(float* __restrict__ d, int stride) {
  __shared__ float line[4][64];
  int i  = threadIdx.x & 63;
  int li = threadIdx.x >> 6;
  int l  = blockIdx.x * 4 + li;
  int base = (l / stride) * (stride * 64) + (l % stride);
  int idx  = base + i * stride;
  line[li][i] = d[idx];
  __syncthreads();
  float fi   = (float)i;
  float best = line[li][i];            // j == i candidate (cost 0)
  #pragma unroll 8
  for (int j = 0; j < 64; ++j) {
    float diff = fi - (float)j;
    best = fminf(best, line[li][j] + diff * diff);
  }
  d[idx] = best;                       // lines are disjoint per block: safe
}

// ---------------------------------------------------------- fused reduction
// Partial slots: 0 focal | 1-3 inter_c | 4-6 psum_c | 7-9 tsum_c |
// 10 skel_sum | 11 pred*skel | 12-15 predsum_b | 16-19 term_b | 20-23 gcnt_b
__global__ void reduce_kernel(const float* __restrict__ logits,
                              const long long* __restrict__ tgt,
                              const int* __restrict__ skel,
                              const float* __restrict__ pos,
                              const float* __restrict__ neg,
                              float* __restrict__ partials) {
  float acc[26];
  #pragma unroll
  for (int q = 0; q < 26; ++q) acc[q] = 0.0f;

  int tid  = threadIdx.x;
  int gid  = blockIdx.x * blockDim.x + tid;
  int nthr = gridDim.x * blockDim.x;           // 65536

  #pragma unroll
  for (int b = 0; b < 4; ++b) {                // unrolled -> static acc idx
    const float*     lg  = logits + (size_t)b * 3 * NVOX;
    const long long* tb  = tgt    + (size_t)b * NVOX;
    const int*       skb = skel   + (size_t)b * NVOX;
    const float*     pb  = pos    + (size_t)b * NVOX;
    const float*     nb  = neg    + (size_t)b * NVOX;
    for (int v = gid; v < NVOX; v += nthr) {
      float x0 = lg[v], x1 = lg[v + NVOX], x2 = lg[v + 2 * NVOX];
      __builtin_prefetch(lg + v + nthr, 0, 1);       // global_prefetch_b8
      float m  = fmaxf(x0, fmaxf(x1, x2));
      float e0 = expf(x0 - m), e1 = expf(x1 - m), e2 = expf(x2 - m);
      float s  = e0 + e1 + e2;
      float ls = logf(s);
      float p0 = e0 / s, p1 = e1 / s, p2 = e2 / s;

      int   t     = (int)tb[v];
      float ptv   = (t == 0) ? p0 : ((t == 1) ? p1 : p2);
      float xt    = (t == 0) ? x0 : ((t == 1) ? x1 : x2);
      float logpt = (xt - m) - ls;
      float alpha = (t == 0) ? 0.3f : ((t == 1) ? 3.0f : 0.3f);
      float om    = 1.0f - ptv;
      acc[0] -= alpha * om * om * logpt;             // focal sum

      if (t == 0) { acc[1] += p0; acc[7] += 1.0f; }
      if (t == 1) { acc[2] += p1; acc[8] += 1.0f; }
      if (t == 2) { acc[3] += p2; acc[9] += 1.0f; }
      acc[4] += p0; acc[5] += p1; acc[6] += p2;

      float sk = (float)skb[v];
      acc[10] += sk;
      acc[11] += p1 * sk;

      float sg = sqrtf(pb[v]) - sqrtf(nb[v]);        // signed distance
      acc[12 + b] += p1;
      acc[16 + b] += p1 * sg;
      acc[20 + b] += (t == 1) ? 1.0f : 0.0f;
    }
  }

  // deterministic block tree reduction
  __shared__ float red[256];
  float* outp = partials + (size_t)blockIdx.x * 32;
  for (int q = 0; q < 26; ++q) {
    red[tid] = acc[q];
    __syncthreads();
    for (int s = 128; s > 0; s >>= 1) {
      if (tid < s) red[tid] += red[tid + s];
      __syncthreads();
    }
    if (tid == 0) outp[q] = red[0];
    __syncthreads();
  }
  if (tid == 0) {
    for (int q = 26; q < 32; ++q) outp[q] = 0.0f;    // pad rows for WMMA
  }
}

// ----------------------------------------------------------------- finalize
// Sum partials[256][32] over dim 0 with chained V_WMMA_F32_16X16X4_F32:
// A[M][K] = partials[4*ch + K][M(+16g)], B = ones(4x16)  =>  D[M][*] += rowsum.
// A layout (ISA 7.12.2): lanes 0-15 hold K0/K1, lanes 16-31 hold K2/K3.
// D layout: column N=0 rows 0-7 in lane 0, rows 8-15 in lane 16.
__global__ void finalize_kernel(const float* __restrict__ partials,
                                float* __restrict__ out) {
  int lane = threadIdx.x;                  // 32 threads, EXEC all-ones
  int q0   = lane & 15;
  int koff = (lane < 16) ? 0 : 2;
  v2f ones; ones.x = 1.0f; ones.y = 1.0f;
  __shared__ float sums[32];

  for (int g = 0; g < 2; ++g) {
    int q = q0 + g * 16;
    v8f c = {0.0f, 0.0f, 0.0f, 0.0f, 0.0f, 0.0f, 0.0f, 0.0f};
    for (int ch = 0; ch < 64; ++ch) {
      int p = ch * 4 + koff;
      v2f a;
      a.x = partials[(size_t)p * 32 + q];
      a.y = partials[(size_t)(p + 1) * 32 + q];
      c = __builtin_amdgcn_wmma_f32_16x16x4_f32(
              /*neg_a=*/false, a, /*neg_b=*/false, ones,
              /*c_mod=*/(short)0, c, /*reuse_a=*/false, /*reuse_b=*/false);
    }
    if (lane == 0) {
      #pragma unroll
      for (int r = 0; r < 8; ++r) sums[g * 16 + r] = c[r];
    }
    if (lane == 16) {
      #pragma unroll
      for (int r = 0; r < 8; ++r) sums[g * 16 + 8 + r] = c[r];
    }
    __syncthreads();
  }

  if (lane == 0) {
    const float TVf = (float)TVOX, NVf = (float)NVOX;
    float focal = sums[0] / TVf;
    float cw[3] = {0.3f, 3.0f, 0.3f};
    float dice = 0.0f;
    for (int ci = 0; ci < 3; ++ci) {
      float inter = sums[1 + ci];
      float un    = sums[4 + ci] + sums[7 + ci];
      dice += cw[ci] * (1.0f - (2.0f * inter + 1.0f) / (un + 1.0f));
    }
    float lfd = focal + dice / 3.6f;
    float lsk = 1.0f - (sums[11] + 1.0f) / (sums[10] + 1.0f);
    float lb = 0.0f;
    for (int b = 0; b < 4; ++b) {
      float meanp = sums[12 + b] / NVf;
      float term  = sums[16 + b] / NVf;
      float gsum  = sums[20 + b];
      float pb = (gsum == 0.0f) ? meanp
               : ((gsum == NVf) ? (1.0f - meanp) : term);
      lb += pb;
    }
    lb *= 0.25f;
    out[0] = 0.3f * lfd + 0.3f * lsk + 0.2f * lb;
    out[1] = lfd;
    out[2] = lsk;
    out[3] = lb;
  }
}

// -------------------------------------------------------------------- launch
extern "C" void kernel_launch(void* const* d_in, const int* in_sizes, int n_in,
                              void* d_out, int out_size, void* d_ws, size_t ws_size,
                              hipStream_t stream) {
  (void)in_sizes; (void)n_in; (void)out_size; (void)ws_size;
  const float*     logits = (const float*)d_in[0];
  const long long* tgt    = (const long long*)d_in[1];
  const int*       skel   = (const int*)d_in[2];
  float* out = (float*)d_out;

  float* pos      = (float*)d_ws;          // TVOX floats (4 MB)
  float* neg      = pos + TVOX;            // TVOX floats (4 MB)
  float* partials = neg + TVOX;            // NBLK*32 floats (32 KB)

  edt_init_kernel<<<TVOX / 256, 256, 0, stream>>>(tgt, pos, neg);

  const int strides[3] = {4096, 64, 1};    // D, H, W (reference axis order)
  for (int k = 0; k < 3; ++k) {
    edt_pass_kernel<<<4096, 256, 0, stream>>>(pos, strides[k]);
    edt_pass_kernel<<<4096, 256, 0, stream>>>(neg, strides[k]);
  }

  reduce_kernel<<<NBLK, 256, 0, stream>>>(logits, tgt, skel, pos, neg, partials);
  finalize_kernel<<<1, 32, 0, stream>>>(partials, out);
}